// GATLayer_77910706749978
// MI455X (gfx1250) — compile-verified
//
#include <hip/hip_runtime.h>
#include <hip/hip_bf16.h>

#define F_DIM 64
#define NEG_SLOPE 0.01f

typedef __attribute__((ext_vector_type(2))) float v2f;
typedef __attribute__((ext_vector_type(8))) float v8f;

// ---------------------------------------------------------------------------
// Kernel 1: per-node attention scalars via V_WMMA_F32_16X16X4_F32.
// One wave handles 16 nodes. D = sum_c A_c x B_c over 16 K-chunks of 4:
//   A_c[i][k] = h[base+i][4c+k]           (16x4, f32)
//   B_c[k][j] = w1[4c+k] (j<8) or w2[4c+k] (j>=8)
// => D[i][0] = a1[base+i], D[i][8] = a2[base+i]  (exact f32 dot products)
// ---------------------------------------------------------------------------
__global__ void gat_node_scores_wmma(const float* __restrict__ h,
                                     const float* __restrict__ attn_w, // [128]
                                     float* __restrict__ a1,
                                     float* __restrict__ a2,
                                     int N) {
    const int wave = (blockIdx.x * blockDim.x + threadIdx.x) >> 5;
    const int lane = threadIdx.x & 31;
    const int base = wave * 16;
    if (base >= N) return;               // wave-uniform: whole wave exits, EXEC stays all-1s otherwise

    // A-matrix lane mapping (ISA 7.12.2, 32-bit A 16x4):
    //   lanes 0-15: M=lane, VGPR0=K0, VGPR1=K1 ; lanes 16-31: M=lane-16, K2/K3
    const int mrow  = lane & 15;
    const int khalf = lane >> 4;                       // 0 -> K{0,1}, 1 -> K{2,3}
    const int node  = min(base + mrow, N - 1);         // clamp: keep loads in-bounds, no divergence
    const float* __restrict__ hrow = h + (size_t)node * F_DIM;

    // B-matrix: column j = lane&15 selects w1 (j<8) or w2 (j>=8); K mapping mirrors A.
    const float* __restrict__ wsel = attn_w + ((mrow < 8) ? 0 : F_DIM);

    v8f d = {};
#pragma unroll
    for (int c = 0; c < 16; ++c) {
        v2f A, B;
        A.x = hrow[4 * c + 2 * khalf + 0];
        A.y = hrow[4 * c + 2 * khalf + 1];
        B.x = wsel[4 * c + 2 * khalf + 0];
        B.y = wsel[4 * c + 2 * khalf + 1];
        // 8 args: (neg_a, A, neg_b, B, c_mod, C, reuse_a, reuse_b)
        d = __builtin_amdgcn_wmma_f32_16x16x4_f32(false, A, false, B,
                                                  (short)0, d, false, false);
    }

    // D layout (ISA 7.12.2): VGPR v, lanes 0-15 -> M=v, N=lane ; lanes 16-31 -> M=8+v, N=lane-16.
    // a1 lives in column N=0 (lanes 0 and 16), a2 in column N=8 (lanes 8 and 24).
    float* dstp = nullptr;
    int nb = 0;
    if (lane == 0)  { dstp = a1; nb = base; }
    if (lane == 16) { dstp = a1; nb = base + 8; }
    if (lane == 8)  { dstp = a2; nb = base; }
    if (lane == 24) { dstp = a2; nb = base + 8; }
    if (dstp) {
#pragma unroll
        for (int v = 0; v < 8; ++v)
            if (nb + v < N) dstp[nb + v] = d[v];
    }
}

// ---------------------------------------------------------------------------
// Kernel 2: row_ptr[n] = lower_bound(dst, n) over the sorted dst array.
// Contiguous segment for node n is [row_ptr[n], row_ptr[n+1]).
// ---------------------------------------------------------------------------
__global__ void gat_rowptr(const int* __restrict__ dst, int E,
                           int* __restrict__ rowptr, int N) {
    const int n = blockIdx.x * blockDim.x + threadIdx.x;
    if (n > N) return;
    int lo = 0, hi = E;
    while (lo < hi) {
        const int mid = (lo + hi) >> 1;
        if (dst[mid] < n) lo = mid + 1; else hi = mid;
    }
    rowptr[n] = lo;
}

// ---------------------------------------------------------------------------
// Kernel 3: one wave32 per destination node. Pass 1: wave-parallel max of
// leaky-relu logits. Pass 2 (fused): denom = sum exp(e-m) and
// acc = sum exp(e-m)*h[src]; normalize once at the end. 2 floats/lane = F=64.
// All h-row gathers hit the 192MB L2 (h is 12.8MB).
// ---------------------------------------------------------------------------
__global__ void gat_segment_softmax_scatter(const float* __restrict__ h,
                                            const float* __restrict__ a1,
                                            const float* __restrict__ a2,
                                            const int* __restrict__ src,
                                            const int* __restrict__ rowptr,
                                            float* __restrict__ out,
                                            int N) {
    const int n    = (blockIdx.x * blockDim.x + threadIdx.x) >> 5;
    const int lane = threadIdx.x & 31;
    if (n >= N) return;

    const int beg = rowptr[n];
    const int end = rowptr[n + 1];
    float2 acc = {0.f, 0.f};

    if (beg >= end) {                       // empty segment -> segment_sum gives 0
        ((float2*)(out + (size_t)n * F_DIM))[lane] = acc;
        return;
    }

    const float a2n = a2[n];

    // pass 1: segment max (lanes stride the edge list, then wave-reduce)
    float m = -__builtin_inff();
    for (int k = beg + lane; k < end; k += 32) {
        float e = a1[src[k]] + a2n;
        e = (e >= 0.f) ? e : NEG_SLOPE * e;
        m = fmaxf(m, e);
    }
#pragma unroll
    for (int off = 16; off > 0; off >>= 1)
        m = fmaxf(m, __shfl_xor(m, off, 32));

    // pass 2: fused exp-weights + weighted feature accumulation
    float denom = 0.f;
    for (int blk = beg; blk < end; blk += 32) {
        const int k = blk + lane;
        float wgt = 0.f;
        int   s   = 0;
        if (k < end) {
            s = src[k];
            float e = a1[s] + a2n;
            e = (e >= 0.f) ? e : NEG_SLOPE * e;
            wgt = __expf(e - m);
        }
        const int cnt = min(32, end - blk);
        for (int j = 0; j < cnt; ++j) {
            const float wj = __shfl(wgt, j, 32);   // broadcast edge j's weight
            const int   sj = __shfl(s,   j, 32);   // broadcast edge j's src node
            const float2 v = ((const float2*)(h + (size_t)sj * F_DIM))[lane];
            acc.x = fmaf(wj, v.x, acc.x);
            acc.y = fmaf(wj, v.y, acc.y);
            denom += wj;                           // identical across lanes
        }
    }

    const float inv = 1.f / denom;
    float2 o = {acc.x * inv, acc.y * inv};
    ((float2*)(out + (size_t)n * F_DIM))[lane] = o;
}

// ---------------------------------------------------------------------------
extern "C" void kernel_launch(void* const* d_in, const int* in_sizes, int n_in,
                              void* d_out, int out_size, void* d_ws, size_t ws_size,
                              hipStream_t stream) {
    const float* h      = (const float*)d_in[0];   // [N, 64]
    const float* attn_w = (const float*)d_in[1];   // [128]
    const int*   src    = (const int*)d_in[2];     // [E]
    const int*   dst    = (const int*)d_in[3];     // [E] sorted
    const int N = in_sizes[0] / F_DIM;
    const int E = in_sizes[2];
    float* out = (float*)d_out;

    // workspace: a1[N] f32 | a2[N] f32 | rowptr[N+1] i32  (~600 KB)
    float* a1     = (float*)d_ws;
    float* a2     = a1 + N;
    int*   rowptr = (int*)(a2 + N);

    // Kernel 1: WMMA node scores. 16 nodes/wave, 8 waves/block.
    {
        const int tiles  = (N + 15) / 16;
        const int blocks = (tiles + 7) / 8;
        gat_node_scores_wmma<<<blocks, 256, 0, stream>>>(h, attn_w, a1, a2, N);
    }
    // Kernel 2: segment boundaries (N+1 threads).
    gat_rowptr<<<(N + 256) / 256, 256, 0, stream>>>(dst, E, rowptr, N);
    // Kernel 3: one wave per node, 8 waves/block.
    {
        const int blocks = (N + 7) / 8;
        gat_segment_softmax_scatter<<<blocks, 256, 0, stream>>>(h, a1, a2, src,
                                                                rowptr, out, N);
    }
}